// DMoN_60266981097667
// MI455X (gfx1250) — compile-verified
//
#include <hip/hip_runtime.h>
#include <hip/hip_bf16.h>
#include <math.h>

typedef float v2f __attribute__((ext_vector_type(2)));
typedef float v8f __attribute__((ext_vector_type(8)));

// ---------------------------------------------------------------------------
// D[M,N] = A[M,K] @ B[K,N], row-major fp32, M%16==0, N%32==0, K%4==0.
// One wave computes a 16x32 tile (two 16x16 WMMA accumulators sharing the A
// fragment -> half the A loads, two independent WMMA chains for latency
// hiding).  V_WMMA_F32_16X16X4_F32 layouts per ISA 7.12.2:
//   A (16x4):  lanes 0-15 row m0+l, {v0,v1}={K,K+1}; lanes 16-31 {K+2,K+3}
//   B (4x16):  v0/v1 = rows K(/+1) lanes 0-15, rows K+2(/+3) lanes 16-31
//   C/D:       VGPR g -> rows g (lanes 0-15) and g+8 (lanes 16-31)
// ---------------------------------------------------------------------------
__global__ __launch_bounds__(32)
void gemm_wmma_f32(const float* __restrict__ A, const float* __restrict__ B,
                   float* __restrict__ D, int M, int K, int N) {
  const int lane = threadIdx.x & 31;
  const int half = lane >> 4;          // 0: K pair {0,1}; 1: K pair {2,3}
  const int l16  = lane & 15;
  const int m0 = blockIdx.x * 16;
  const int n0 = blockIdx.y * 32;

  const float* arow = A + (size_t)(m0 + l16) * K + 2 * half;
  const float* bp   = B + (size_t)(2 * half) * N + n0 + l16;      // n-tile 0
  const float* bq   = bp + 16;                                    // n-tile 1

  v8f acc0 = {}, acc1 = {};
  for (int k0 = 0; k0 < K; k0 += 4) {
    v2f a, p, q;
    a.x = arow[k0];
    a.y = arow[k0 + 1];
    p.x = bp[(size_t)k0 * N];
    p.y = bp[(size_t)(k0 + 1) * N];
    q.x = bq[(size_t)k0 * N];
    q.y = bq[(size_t)(k0 + 1) * N];
    acc0 = __builtin_amdgcn_wmma_f32_16x16x4_f32(false, a, false, p,
                                                 (short)0, acc0, false, false);
    acc1 = __builtin_amdgcn_wmma_f32_16x16x4_f32(false, a, false, q,
                                                 (short)0, acc1, false, false);
  }
  float* drow = D + (size_t)(m0 + 8 * half) * N + n0 + l16;
#pragma unroll
  for (int g = 0; g < 8; ++g) {
    drow[(size_t)g * N]      = acc0[g];
    drow[(size_t)g * N + 16] = acc1[g];
  }
}

// ---------------------------------------------------------------------------
// GCN normalization helpers
// ---------------------------------------------------------------------------
__global__ void col_deg_kernel(const long long* __restrict__ ei,
                               const float* __restrict__ ew,
                               float* __restrict__ deg, int E) {
  int t = blockIdx.x * blockDim.x + threadIdx.x;
  int stride = gridDim.x * blockDim.x;
  for (int e = t; e < E; e += stride)
    atomicAdd(&deg[ei[(size_t)E + e]], ew[e]);   // segment_sum over col
}

__global__ void make_dinv_kernel(float* __restrict__ deg, int n) {
  int i = blockIdx.x * blockDim.x + threadIdx.x;
  if (i < n) {
    float d = deg[i] + 1.0f;                     // + self-loop weight 1
    deg[i] = (d > 0.f) ? rsqrtf(d) : 0.f;        // in-place -> dinv
  }
}

// agg[i,:] = dinv[i]^2 * xw[i,:]   (self-loop contribution, also zero-inits)
// One thread per float4; totalv = N*F/4, Fv = F/4.
__global__ void selfloop_init_kernel(const float4* __restrict__ xw,
                                     const float* __restrict__ dinv,
                                     float4* __restrict__ agg,
                                     size_t totalv, int Fv) {
  size_t idx = (size_t)blockIdx.x * blockDim.x + threadIdx.x;
  size_t stride = (size_t)gridDim.x * blockDim.x;
  for (size_t t = idx; t < totalv; t += stride) {
    float di = dinv[t / Fv];
    float c = di * di;
    float4 v = xw[t];
    agg[t] = make_float4(c * v.x, c * v.y, c * v.z, c * v.w);
  }
}

// agg[col,:] += dinv[row]*ew*dinv[col] * xw[row,:]
// One thread per (edge, 4 features): b128 load, 4 scalar atomics.
__global__ void edge_agg_kernel(const long long* __restrict__ ei,
                                const float* __restrict__ ew,
                                const float* __restrict__ dinv,
                                const float* __restrict__ xw,
                                float* __restrict__ agg, int E, int F) {
  const int Fv = F >> 2;
  size_t idx = (size_t)blockIdx.x * blockDim.x + threadIdx.x;
  size_t stride = (size_t)gridDim.x * blockDim.x;
  size_t total = (size_t)E * Fv;
  for (size_t t = idx; t < total; t += stride) {
    size_t e = t / Fv;
    int j = (int)(t - e * Fv) << 2;
    long long r = ei[e];
    long long c = ei[(size_t)E + e];
    float coeff = dinv[r] * ew[e] * dinv[c];
    float4 v = *(const float4*)(xw + (size_t)r * F + j);
    float* dst = agg + (size_t)c * F + j;
    atomicAdd(dst + 0, coeff * v.x);
    atomicAdd(dst + 1, coeff * v.y);
    atomicAdd(dst + 2, coeff * v.z);
    atomicAdd(dst + 3, coeff * v.w);
  }
}

__global__ void bias_relu_kernel(float4* __restrict__ h,
                                 const float4* __restrict__ b,
                                 size_t totalv, int Fv) {
  size_t idx = (size_t)blockIdx.x * blockDim.x + threadIdx.x;
  size_t stride = (size_t)gridDim.x * blockDim.x;
  for (size_t t = idx; t < totalv; t += stride) {
    float4 v = h[t];
    float4 bb = b[t % Fv];
    h[t] = make_float4(fmaxf(v.x + bb.x, 0.f), fmaxf(v.y + bb.y, 0.f),
                       fmaxf(v.z + bb.z, 0.f), fmaxf(v.w + bb.w, 0.f));
  }
}

// ---------------------------------------------------------------------------
// Softmax over C=64 columns, one block (2 waves) per row.
// ---------------------------------------------------------------------------
__global__ void softmax_rows_kernel(const float* __restrict__ logits,
                                    const float* __restrict__ bias,
                                    float* __restrict__ s, int C) {
  __shared__ float sh[64];
  int i = blockIdx.x, k = threadIdx.x;
  float v = logits[(size_t)i * C + k] + bias[k];
  sh[k] = v; __syncthreads();
  for (int off = C >> 1; off > 0; off >>= 1) {
    if (k < off) sh[k] = fmaxf(sh[k], sh[k + off]);
    __syncthreads();
  }
  float mx = sh[0]; __syncthreads();
  float e = expf(v - mx);
  sh[k] = e; __syncthreads();
  for (int off = C >> 1; off > 0; off >>= 1) {
    if (k < off) sh[k] += sh[k + off];
    __syncthreads();
  }
  s[(size_t)i * C + k] = e / sh[0];
}

// ---------------------------------------------------------------------------
// Loss pieces
// ---------------------------------------------------------------------------
// degrees[row] += ew; sumw += ew  (sum of all adj entries == 2m)
__global__ void row_deg_sum_kernel(const long long* __restrict__ ei,
                                   const float* __restrict__ ew,
                                   float* __restrict__ degrees,
                                   float* __restrict__ sumw, int E) {
  int t = blockIdx.x * blockDim.x + threadIdx.x;
  int stride = gridDim.x * blockDim.x;
  float local = 0.f;
  for (int e = t; e < E; e += stride) {
    float w = ew[e];
    atomicAdd(&degrees[ei[e]], w);
    local += w;
  }
  for (int off = 16; off > 0; off >>= 1) local += __shfl_down(local, off, 32);
  if ((threadIdx.x & 31) == 0) atomicAdd(sumw, local);
}

// adjs[row,k..k+3] += ew * s[col,k..k+3]   (sparse adj @ s, float4 per thread)
__global__ void adj_s_scatter_kernel(const long long* __restrict__ ei,
                                     const float* __restrict__ ew,
                                     const float* __restrict__ s,
                                     float* __restrict__ adjs, int E, int C) {
  const int Cv = C >> 2;
  size_t idx = (size_t)blockIdx.x * blockDim.x + threadIdx.x;
  size_t stride = (size_t)gridDim.x * blockDim.x;
  size_t total = (size_t)E * Cv;
  for (size_t t = idx; t < total; t += stride) {
    size_t e = t / Cv;
    int k = (int)(t - e * Cv) << 2;
    long long r = ei[e];
    long long c = ei[(size_t)E + e];
    float w = ew[e];
    float4 v = *(const float4*)(s + (size_t)c * C + k);
    float* dst = adjs + (size_t)r * C + k;
    atomicAdd(dst + 0, w * v.x);
    atomicAdd(dst + 1, w * v.y);
    atomicAdd(dst + 2, w * v.z);
    atomicAdd(dst + 3, w * v.w);
  }
}

// out += sum(a[i]*b[i]) over nv float4 chunks
__global__ void dot_reduce_kernel(const float4* __restrict__ a,
                                  const float4* __restrict__ b,
                                  float* __restrict__ out, size_t nv) {
  size_t idx = (size_t)blockIdx.x * blockDim.x + threadIdx.x;
  size_t stride = (size_t)gridDim.x * blockDim.x;
  float local = 0.f;
  for (size_t t = idx; t < nv; t += stride) {
    float4 av = a[t], bv = b[t];
    local += av.x * bv.x + av.y * bv.y + av.z * bv.z + av.w * bv.w;
  }
  for (int off = 16; off > 0; off >>= 1) local += __shfl_down(local, off, 32);
  if ((threadIdx.x & 31) == 0) atomicAdd(out, local);
}

// ca[k] += sum_i s[i,k]*degrees[i]; cs[k] += sum_i s[i,k].  blockDim == C.
__global__ void ca_csize_kernel(const float* __restrict__ s,
                                const float* __restrict__ degrees,
                                float* __restrict__ ca, float* __restrict__ cs,
                                int n, int C) {
  int k = threadIdx.x;
  float lca = 0.f, lcs = 0.f;
  for (int i = blockIdx.x; i < n; i += gridDim.x) {
    float sv = s[(size_t)i * C + k];
    lca += sv * degrees[i];
    lcs += sv;
  }
  atomicAdd(&ca[k], lca);
  atomicAdd(&cs[k], lcs);
}

__global__ void finalize_kernel(const float* __restrict__ scal,   // [sumw, trace]
                                const float* __restrict__ ca,
                                const float* __restrict__ cs,
                                float* __restrict__ out, float n, int C) {
  float twom = scal[0];       // degrees.sum() == 2m
  float trace = scal[1];      // tr(s^T A s)
  float dot = 0.f, norm2 = 0.f;
  for (int k = 0; k < C; ++k) {
    dot += ca[k] * ca[k];
    norm2 += cs[k] * cs[k];
  }
  float trace_norm = dot / twom;                  // (ca.ca)/(2m)
  float spectral = -(trace - trace_norm) / twom;  // /(2m)
  float cluster = sqrtf(norm2) / n * sqrtf((float)C) - 1.f;
  out[0] = 100.f * (spectral + cluster);
  out[1] = 100.f * spectral;
  out[2] = 100.f * cluster;
}

// ---------------------------------------------------------------------------
extern "C" void kernel_launch(void* const* d_in, const int* in_sizes, int n_in,
                              void* d_out, int out_size, void* d_ws, size_t ws_size,
                              hipStream_t stream) {
  (void)n_in; (void)out_size; (void)ws_size;
  const float*     x  = (const float*)d_in[0];
  const float*     ea = (const float*)d_in[1];
  const float*     W1 = (const float*)d_in[2];
  const float*     b1 = (const float*)d_in[3];
  const float*     W2 = (const float*)d_in[4];
  const float*     b2 = (const float*)d_in[5];
  const float*     Wm = (const float*)d_in[6];
  const float*     bm = (const float*)d_in[7];
  const long long* ei = (const long long*)d_in[8];   // int64 [2,E]

  const int HID = in_sizes[3];          // 256
  const int C   = in_sizes[7];          // 64
  const int FIN = in_sizes[2] / HID;    // 128
  const int N   = in_sizes[0] / FIN;    // 10000 (multiple of 16)
  const int E   = in_sizes[1];          // 160000

  float* ws      = (float*)d_ws;
  float* dinv    = ws;                                  // N (deg -> dinv)
  float* degrees = dinv + N;                            // N (adj row sums)
  float* bufA    = degrees + N;                         // N*HID (agg1/h1, later adj_s)
  float* bufB    = bufA + (size_t)N * HID;              // N*HID (XW / logits)
  float* bufC    = bufB + (size_t)N * HID;              // N*HID (agg2/h2)
  float* scal    = bufC + (size_t)N * HID;              // [0]=sumw(2m), [1]=trace
  float* ca      = scal + 2;                            // C
  float* cs      = ca + C;                              // C

  float* s_out = (float*)d_out;                         // s occupies first N*C
  const size_t NHv = (size_t)N * HID / 4;               // float4 count
  const int HIDv = HID / 4;

  // --- GCN normalization (same for both layers) ---
  hipMemsetAsync(dinv, 0, (size_t)N * sizeof(float), stream);
  col_deg_kernel<<<256, 256, 0, stream>>>(ei, ea, dinv, E);
  make_dinv_kernel<<<(N + 255) / 256, 256, 0, stream>>>(dinv, N);

  // --- Layer 1: relu(gcn(x, W1) + b1) ---
  gemm_wmma_f32<<<dim3(N / 16, HID / 32), 32, 0, stream>>>(x, W1, bufB, N, FIN, HID);
  selfloop_init_kernel<<<1024, 256, 0, stream>>>((const float4*)bufB, dinv,
                                                 (float4*)bufA, NHv, HIDv);
  edge_agg_kernel<<<2048, 256, 0, stream>>>(ei, ea, dinv, bufB, bufA, E, HID);
  bias_relu_kernel<<<1024, 256, 0, stream>>>((float4*)bufA, (const float4*)b1,
                                             NHv, HIDv);

  // --- Layer 2: relu(gcn(h1, W2) + b2) ---
  gemm_wmma_f32<<<dim3(N / 16, HID / 32), 32, 0, stream>>>(bufA, W2, bufB, N, HID, HID);
  selfloop_init_kernel<<<1024, 256, 0, stream>>>((const float4*)bufB, dinv,
                                                 (float4*)bufC, NHv, HIDv);
  edge_agg_kernel<<<2048, 256, 0, stream>>>(ei, ea, dinv, bufB, bufC, E, HID);
  bias_relu_kernel<<<1024, 256, 0, stream>>>((float4*)bufC, (const float4*)b2,
                                             NHv, HIDv);

  // --- Cluster assignment: s = softmax(h2 @ Wm + bm) ---
  gemm_wmma_f32<<<dim3(N / 16, C / 32), 32, 0, stream>>>(bufC, Wm, bufB, N, HID, C);
  softmax_rows_kernel<<<N, C, 0, stream>>>(bufB, bm, s_out, C);

  // --- Losses ---
  hipMemsetAsync(degrees, 0, (size_t)N * sizeof(float), stream);
  hipMemsetAsync(scal, 0, (size_t)(2 + 2 * C) * sizeof(float), stream);
  hipMemsetAsync(bufA, 0, (size_t)N * C * sizeof(float), stream);   // adj_s

  row_deg_sum_kernel<<<256, 256, 0, stream>>>(ei, ea, degrees, &scal[0], E);
  adj_s_scatter_kernel<<<1024, 256, 0, stream>>>(ei, ea, s_out, bufA, E, C);
  dot_reduce_kernel<<<256, 256, 0, stream>>>((const float4*)bufA,
                                             (const float4*)s_out, &scal[1],
                                             (size_t)N * C / 4);
  ca_csize_kernel<<<256, C, 0, stream>>>(s_out, degrees, ca, cs, N, C);
  finalize_kernel<<<1, 1, 0, stream>>>(scal, ca, cs, s_out + (size_t)N * C, (float)N, C);
}